// MambaSequenceClassifier_79980880986653
// MI455X (gfx1250) — compile-verified
//
#include <hip/hip_runtime.h>
#include <hip/hip_bf16.h>

// ---------------------------------------------------------------------------
// MambaSequenceClassifier forward for MI455X (gfx1250, wave32, WMMA).
// All heavy GEMMs use V_WMMA_F32_16X16X4_F32 (fp32-exact vs reference).
// GEMM pipeline: LDS double-buffered, K-pair (float2) LDS layout so each
// WMMA fragment is a single aligned ds_load_b64 into an even VGPR pair.
// ---------------------------------------------------------------------------

typedef __attribute__((ext_vector_type(2))) float v2f;
typedef __attribute__((ext_vector_type(4))) float v4f;
typedef __attribute__((ext_vector_type(8))) float v8f;

#define D_MODEL   1024
#define D_INNER   2048
#define D_STATE   16
#define DT_RANK   64
#define HIDDEN    512
#define NCLS      16
#define BATCH     8
#define SEQ       4096
#define MROWS     (BATCH * SEQ)          // 32768

enum Epi { EPI_NONE = 0, EPI_BIAS = 1, EPI_BIAS_RELU = 2, EPI_BIAS_SOFTPLUS = 3 };

// ---------------------------------------------------------------------------
// fp32 WMMA GEMM:  C[m,n] = epi( sum_k A[m,k] * W[n,k] + bias[n] )
// A row-major [M x K] (lda), W row-major [N x K] (ldb), C row-major (ldc).
// Block tile 128x128, K-slice 16 (double buffered), 8 waves: 2(M) x 4(N),
// wave tile 64x32 = 4x2 WMMA tiles.  Requires K % 16 == 0.
// ---------------------------------------------------------------------------
template <int EPI>
__global__ __launch_bounds__(256) void gemm_wmma_f32(
    const float* __restrict__ A, const float* __restrict__ W,
    float* __restrict__ C, const float* __restrict__ bias,
    int M, int N, int K, int lda, int ldb, int ldc)
{
    constexpr int BM = 128, BN = 128, BK = 16;
    // K-pair layout: entry [buf][kp][m] = (elem(2*kp), elem(2*kp+1)) of row m.
    __shared__ v2f As[2][BK / 2][BM + 1];
    __shared__ v2f Bs[2][BK / 2][BN + 1];

    const int tid    = threadIdx.x;
    const int lane   = tid & 31;
    const int wave   = tid >> 5;          // 0..7
    const int wm     = wave >> 2;         // 0..1  (M direction)
    const int wn     = wave & 3;          // 0..3  (N direction)
    const int laneLo = lane & 15;
    const bool hi    = lane >= 16;

    const int block_m = blockIdx.x * BM;
    const int block_n = blockIdx.y * BN;

    // Each thread owns rows (row0, row0+64) and K-quad kq of the staged tiles.
    const int row0 = tid >> 2;            // 0..63
    const int kq   = (tid & 3) << 2;      // 0,4,8,12
    const int kp0  = kq >> 1;             // 0,2,4,6

    v8f acc[4][2] = {};                   // 64 accumulator VGPRs
    v4f a0, a1, b0, b1;                   // in-flight K-slice

    auto gload = [&](int k0) {
        const v4f z4 = {0.f, 0.f, 0.f, 0.f};
        int gm0 = block_m + row0;
        int gn0 = block_n + row0;
        a0 = (gm0      < M) ? *(const v4f*)(A + (size_t)gm0        * lda + k0 + kq) : z4;
        a1 = (gm0 + 64 < M) ? *(const v4f*)(A + (size_t)(gm0 + 64) * lda + k0 + kq) : z4;
        b0 = (gn0      < N) ? *(const v4f*)(W + (size_t)gn0        * ldb + k0 + kq) : z4;
        b1 = (gn0 + 64 < N) ? *(const v4f*)(W + (size_t)(gn0 + 64) * ldb + k0 + kq) : z4;
    };
    auto stage = [&](int buf) {
        As[buf][kp0    ][row0]      = (v2f){a0.x, a0.y};
        As[buf][kp0 + 1][row0]      = (v2f){a0.z, a0.w};
        As[buf][kp0    ][row0 + 64] = (v2f){a1.x, a1.y};
        As[buf][kp0 + 1][row0 + 64] = (v2f){a1.z, a1.w};
        Bs[buf][kp0    ][row0]      = (v2f){b0.x, b0.y};
        Bs[buf][kp0 + 1][row0]      = (v2f){b0.z, b0.w};
        Bs[buf][kp0    ][row0 + 64] = (v2f){b1.x, b1.y};
        Bs[buf][kp0 + 1][row0 + 64] = (v2f){b1.z, b1.w};
    };
    auto compute = [&](int buf) {
        #pragma unroll
        for (int kk = 0; kk < BK; kk += 4) {
            // lanes 0-15 use K=(kk,kk+1); lanes 16-31 use K=(kk+2,kk+3)
            const int kp = (kk >> 1) + (hi ? 1 : 0);
            v2f afrag[4], bfrag[2];
            #pragma unroll
            for (int i = 0; i < 4; ++i)
                afrag[i] = As[buf][kp][wm * 64 + i * 16 + laneLo];
            #pragma unroll
            for (int j = 0; j < 2; ++j)
                bfrag[j] = Bs[buf][kp][wn * 32 + j * 16 + laneLo];
            #pragma unroll
            for (int i = 0; i < 4; ++i)
                #pragma unroll
                for (int j = 0; j < 2; ++j)
                    acc[i][j] = __builtin_amdgcn_wmma_f32_16x16x4_f32(
                        false, afrag[i], false, bfrag[j],
                        (short)0, acc[i][j], false, false);
        }
    };

    const int nk = K / BK;
    gload(0);
    stage(0);
    __syncthreads();
    int cur = 0;
    for (int it = 0; it < nk; ++it) {
        if (it + 1 < nk) gload((it + 1) * BK);   // overlap with compute below
        compute(cur);
        if (it + 1 < nk) {
            stage(cur ^ 1);
            __syncthreads();
            cur ^= 1;
        }
    }

    // ---- epilogue + store (C layout: VGPR r -> M=r (lanes 0-15) / r+8 (16-31))
    #pragma unroll
    for (int i = 0; i < 4; ++i) {
        #pragma unroll
        for (int j = 0; j < 2; ++j) {
            int ncol = block_n + wn * 32 + j * 16 + laneLo;
            if (ncol >= N) continue;
            float bv = 0.f;
            if constexpr (EPI != EPI_NONE) bv = bias[ncol];
            #pragma unroll
            for (int r = 0; r < 8; ++r) {
                int mrow = block_m + wm * 64 + i * 16 + r + (hi ? 8 : 0);
                if (mrow >= M) continue;
                float v = acc[i][j][r];
                if constexpr (EPI == EPI_BIAS)           { v += bv; }
                else if constexpr (EPI == EPI_BIAS_RELU) { v += bv; v = fmaxf(v, 0.f); }
                else if constexpr (EPI == EPI_BIAS_SOFTPLUS) {
                    v += bv;
                    v = (v > 20.f) ? v : log1pf(__expf(v));
                }
                C[(size_t)mrow * ldc + ncol] = v;
            }
        }
    }
}

// ---------------------------------------------------------------------------
// Causal depthwise conv1d (width 4) + SiLU.  xz [BL,4096] (xi = first 2048).
// ---------------------------------------------------------------------------
__global__ __launch_bounds__(256) void conv_silu_kernel(
    const float* __restrict__ xz, const float* __restrict__ cw,
    const float* __restrict__ cb, float* __restrict__ xc)
{
    size_t idx = (size_t)blockIdx.x * blockDim.x + threadIdx.x;  // over BL*2048
    int c = (int)(idx & (D_INNER - 1));
    size_t bl = idx >> 11;
    int l = (int)(bl & (SEQ - 1));
    const float* base = xz + bl * (2 * D_INNER) + c;
    float w0 = cw[c * 4 + 0], w1 = cw[c * 4 + 1], w2 = cw[c * 4 + 2], w3 = cw[c * 4 + 3];
    float acc = cb[c] + w3 * base[0];
    if (l >= 1) acc += w2 * base[-(ptrdiff_t)(2 * D_INNER)];
    if (l >= 2) acc += w1 * base[-(ptrdiff_t)(4 * D_INNER)];
    if (l >= 3) acc += w0 * base[-(ptrdiff_t)(6 * D_INNER)];
    xc[idx] = acc / (1.f + __expf(-acc));   // silu
}

// ---------------------------------------------------------------------------
// Selective scan. One lane-quad per (b,d) channel: each lane owns 4 of the 16
// states; y reduced with wave32 shfl_xor butterflies. Fuses D-skip + SiLU(z)
// gate, writes the out_proj input yg [BL, 2048].
// ---------------------------------------------------------------------------
__global__ __launch_bounds__(256) void scan_kernel(
    const float* __restrict__ dt, const float* __restrict__ xc,
    const float* __restrict__ xdbl, const float* __restrict__ xz,
    const float* __restrict__ A_log, const float* __restrict__ Dp,
    float* __restrict__ yg)
{
    int g = blockIdx.x * 256 + threadIdx.x;   // 0 .. 65535
    int q = g & 3;                            // state quad 0..3
    int d = (g >> 2) & (D_INNER - 1);
    int b = g >> 13;                          // /(2048*4)

    float An[4];
    #pragma unroll
    for (int s = 0; s < 4; ++s) An[s] = -__expf(A_log[d * D_STATE + q * 4 + s]);
    const float Dv = Dp[d];

    float h[4] = {0.f, 0.f, 0.f, 0.f};
    for (int l = 0; l < SEQ; ++l) {
        size_t base = (size_t)b * SEQ + l;
        float dtv = dt[base * D_INNER + d];
        float xv  = xc[base * D_INNER + d];
        v4f Bt = *(const v4f*)(xdbl + base * 96 + DT_RANK + q * 4);
        v4f Ct = *(const v4f*)(xdbl + base * 96 + DT_RANK + D_STATE + q * 4);
        float y = 0.f;
        #pragma unroll
        for (int s = 0; s < 4; ++s) {
            float dA = __expf(dtv * An[s]);
            h[s] = dA * h[s] + dtv * Bt[s] * xv;
            y += h[s] * Ct[s];
        }
        y += __shfl_xor(y, 1, 32);
        y += __shfl_xor(y, 2, 32);
        if (q == 0) {
            float zv = xz[base * (2 * D_INNER) + D_INNER + d];
            float sz = zv / (1.f + __expf(-zv));          // silu(z)
            yg[base * D_INNER + d] = (y + xv * Dv) * sz;
        }
    }
}

// ---------------------------------------------------------------------------
// Mean-pool over time (partial sums per 256-l chunk, atomic accumulate).
// ---------------------------------------------------------------------------
__global__ __launch_bounds__(1024) void pool_kernel(
    const float* __restrict__ xm, float* __restrict__ pooled)
{
    int b = blockIdx.x >> 4;
    int chunk = blockIdx.x & 15;
    int d = threadIdx.x;                       // 0..1023
    const float* p = xm + ((size_t)b * SEQ + chunk * 256) * D_MODEL + d;
    float s = 0.f;
    for (int l = 0; l < 256; ++l) s += p[(size_t)l * D_MODEL];
    atomicAdd(&pooled[b * D_MODEL + d], s);
}

__global__ void boundary_kernel(
    const float* __restrict__ pooled, const float* __restrict__ bw,
    const float* __restrict__ bb, float* __restrict__ out)
{
    int t = threadIdx.x;
    if (t >= BATCH * 2) return;
    int b = t >> 1, c = t & 1;
    const float inv = 1.0f / (float)SEQ;
    float acc = bb[c];
    for (int dd = 0; dd < D_MODEL; ++dd)
        acc += pooled[b * D_MODEL + dd] * inv * bw[c * D_MODEL + dd];
    out[b * 2 + c] = 1.f / (1.f + __expf(-acc));
}

// ---------------------------------------------------------------------------
// Launch.  Workspace layout (fp32 elements), total ~1.56 GB:
//   xz[BL*4096] | xc[BL*2048] | xdbl[BL*96] | dt[BL*2048] | yg[BL*2048]
//   | xm[BL*1024] | h1[BL*512] | pooled[8*1024]
// ---------------------------------------------------------------------------
extern "C" void kernel_launch(void* const* d_in, const int* in_sizes, int n_in,
                              void* d_out, int out_size, void* d_ws, size_t ws_size,
                              hipStream_t stream)
{
    const float* x         = (const float*)d_in[0];
    const float* in_proj_w = (const float*)d_in[1];
    const float* conv_w    = (const float*)d_in[2];
    const float* conv_b    = (const float*)d_in[3];
    const float* x_proj_w  = (const float*)d_in[4];
    const float* dt_proj_w = (const float*)d_in[5];
    const float* dt_proj_b = (const float*)d_in[6];
    const float* A_log     = (const float*)d_in[7];
    const float* Dvec      = (const float*)d_in[8];
    const float* out_proj_w= (const float*)d_in[9];
    const float* fc1_w     = (const float*)d_in[10];
    const float* fc1_b     = (const float*)d_in[11];
    const float* out_w     = (const float*)d_in[12];
    const float* out_b     = (const float*)d_in[13];
    const float* bnd_w     = (const float*)d_in[14];
    const float* bnd_b     = (const float*)d_in[15];

    float* ws = (float*)d_ws;
    size_t off = 0;
    float* xz     = ws + off; off += (size_t)MROWS * (2 * D_INNER);
    float* xc     = ws + off; off += (size_t)MROWS * D_INNER;
    float* xdbl   = ws + off; off += (size_t)MROWS * 96;
    float* dt     = ws + off; off += (size_t)MROWS * D_INNER;
    float* yg     = ws + off; off += (size_t)MROWS * D_INNER;
    float* xm     = ws + off; off += (size_t)MROWS * D_MODEL;
    float* h1     = ws + off; off += (size_t)MROWS * HIDDEN;
    float* pooled = ws + off; off += (size_t)BATCH * D_MODEL;

    float* logits     = (float*)d_out;                       // [8,4096,16]
    float* boundaries = (float*)d_out + (size_t)MROWS * NCLS;

    const dim3 blk(256);
    const int MB = MROWS / 128;   // 256 M-blocks

    // 1) in_proj: xz = x @ in_proj_w^T   [32768 x 4096]
    gemm_wmma_f32<EPI_NONE><<<dim3(MB, (2 * D_INNER) / 128), blk, 0, stream>>>(
        x, in_proj_w, xz, nullptr, MROWS, 2 * D_INNER, D_MODEL, D_MODEL, D_MODEL, 2 * D_INNER);

    // 2) causal depthwise conv + SiLU -> xc
    conv_silu_kernel<<<(int)(((size_t)MROWS * D_INNER) / 256), blk, 0, stream>>>(
        xz, conv_w, conv_b, xc);

    // 3) x_proj: xdbl = xc @ x_proj_w^T   [32768 x 96]
    gemm_wmma_f32<EPI_NONE><<<dim3(MB, 1), blk, 0, stream>>>(
        xc, x_proj_w, xdbl, nullptr, MROWS, DT_RANK + 2 * D_STATE, D_INNER, D_INNER, D_INNER, 96);

    // 4) dt_proj + softplus: dt = softplus(xdbl[:, :64] @ dt_proj_w^T + b)
    gemm_wmma_f32<EPI_BIAS_SOFTPLUS><<<dim3(MB, D_INNER / 128), blk, 0, stream>>>(
        xdbl, dt_proj_w, dt, dt_proj_b, MROWS, D_INNER, DT_RANK, 96, DT_RANK, D_INNER);

    // 5) selective scan (+ D-skip + SiLU(z) gate) -> yg
    scan_kernel<<<(BATCH * D_INNER * 4) / 256, blk, 0, stream>>>(
        dt, xc, xdbl, xz, A_log, Dvec, yg);

    // 6) out_proj: xm = yg @ out_proj_w^T   [32768 x 1024]
    gemm_wmma_f32<EPI_NONE><<<dim3(MB, D_MODEL / 128), blk, 0, stream>>>(
        yg, out_proj_w, xm, nullptr, MROWS, D_MODEL, D_INNER, D_INNER, D_INNER, D_MODEL);

    // 7) fc1 + relu: h1 = relu(xm @ fc1_w^T + b)
    gemm_wmma_f32<EPI_BIAS_RELU><<<dim3(MB, HIDDEN / 128), blk, 0, stream>>>(
        xm, fc1_w, h1, fc1_b, MROWS, HIDDEN, D_MODEL, D_MODEL, D_MODEL, HIDDEN);

    // 8) head: logits = h1 @ out_w^T + b   [32768 x 16]
    gemm_wmma_f32<EPI_BIAS><<<dim3(MB, 1), blk, 0, stream>>>(
        h1, out_w, logits, out_b, MROWS, NCLS, HIDDEN, HIDDEN, HIDDEN, NCLS);

    // 9) mean-pool over time + boundary head
    hipMemsetAsync(pooled, 0, (size_t)BATCH * D_MODEL * sizeof(float), stream);
    pool_kernel<<<BATCH * 16, dim3(1024), 0, stream>>>(xm, pooled);
    boundary_kernel<<<1, 32, 0, stream>>>(pooled, bnd_w, bnd_b, boundaries);
}